// GIN_2997887173234
// MI455X (gfx1250) — compile-verified
//
#include <hip/hip_runtime.h>

#define NODES   100000
#define DIM     64
#define NLAYERS 3
#define BN_EPS  1e-5f

typedef __attribute__((ext_vector_type(2))) float v2f;
typedef __attribute__((ext_vector_type(8))) float v8f;

// ---------------------------------------------------------------------------
// agg[n,d] = (1 + eps[l]) * h[n,d]       (float4 vectorized, initializes the
// scatter-sum accumulator so no separate zeroing pass is needed)
// ---------------------------------------------------------------------------
__global__ __launch_bounds__(256) void k_combine(const float* __restrict__ h,
                                                 const float* __restrict__ eps, int l,
                                                 float* __restrict__ agg, int n4) {
    int i = blockIdx.x * 256 + threadIdx.x;
    if (i >= n4) return;
    float s = 1.0f + eps[l];
    float4 v = ((const float4*)h)[i];
    v.x *= s; v.y *= s; v.z *= s; v.w *= s;
    ((float4*)agg)[i] = v;
}

// ---------------------------------------------------------------------------
// One wave per edge: agg[dst] += edge_w * h[src]   (coalesced float2 gather,
// 64 global f32 atomic adds per edge; L2-resident target buffer)
// ---------------------------------------------------------------------------
__global__ __launch_bounds__(256) void k_scatter(const float* __restrict__ h,
                                                 const float* __restrict__ ew,
                                                 const int* __restrict__ esrc,
                                                 const int* __restrict__ edst,
                                                 float* __restrict__ agg, int E) {
    int tid  = blockIdx.x * 256 + threadIdx.x;
    int e    = tid >> 5;
    int lane = tid & 31;
    if (e >= E) return;
    int   s = esrc[e];
    int   d = edst[e];
    float w = ew[e];
    float2 v = *(const float2*)&h[s * DIM + lane * 2];
    float* dst = &agg[d * DIM + lane * 2];
    atomicAdd(dst + 0, w * v.x);
    atomicAdd(dst + 1, w * v.y);
}

__global__ void k_zero(float* __restrict__ p, int n) {
    int i = blockIdx.x * blockDim.x + threadIdx.x;
    if (i < n) p[i] = 0.0f;
}

// ---------------------------------------------------------------------------
// WMMA f32 GEMM:  Y = op(A) @ W + bias,  plus per-column sum / sum-of-squares
// accumulation for the following BatchNorm. op(A) = A, or relu(A*scale+shift)
// (fused BN+ReLU of the previous stage) when BN_ON_A is set.
// One wave computes a 16x64 output slab using V_WMMA_F32_16X16X4_F32.
// ---------------------------------------------------------------------------
template <bool BN_ON_A>
__global__ __launch_bounds__(256) void k_gemm_wmma(
        const float* __restrict__ A,        // [N,64]
        const float* __restrict__ W,        // [64,64] row = k, col = out
        const float* __restrict__ bias,     // [64]
        const float* __restrict__ bn_scale, // [64] (BN_ON_A only)
        const float* __restrict__ bn_shift, // [64] (BN_ON_A only)
        float* __restrict__ Y,              // [N,64]
        float* __restrict__ st_sum,         // [64]
        float* __restrict__ st_sq,          // [64]
        int row_tiles) {
    __shared__ float ls[128];               // block-level BN-stat pre-reduction
    int t = threadIdx.x;
    if (t < 128) ls[t] = 0.0f;
    __syncthreads();

    int wave = blockIdx.x * 8 + (t >> 5);   // 8 waves / block
    int lane = t & 31;
    int half = lane >> 4;                   // which 16-lane half
    int mrow = lane & 15;

    if (wave < row_tiles) {                 // wave-uniform guard: EXEC all-1s inside
        const float* arow = A + (wave * 16 + mrow) * DIM;
        v8f acc[4] = {v8f{}, v8f{}, v8f{}, v8f{}};

        #pragma unroll
        for (int k0 = 0; k0 < DIM; k0 += 4) {
            int ka = k0 + 2 * half;         // this lane's K pair
            float2 av = *(const float2*)&arow[ka];
            if (BN_ON_A) {
                float2 sc = *(const float2*)&bn_scale[ka];
                float2 sh = *(const float2*)&bn_shift[ka];
                av.x = fmaxf(av.x * sc.x + sh.x, 0.0f);
                av.y = fmaxf(av.y * sc.y + sh.y, 0.0f);
            }
            v2f a; a.x = av.x; a.y = av.y;
            const float* w0 = W + ka * DIM + mrow;
            #pragma unroll
            for (int tc = 0; tc < 4; ++tc) {   // 4 column tiles of 16
                v2f b; b.x = w0[tc * 16]; b.y = w0[DIM + tc * 16];
                acc[tc] = __builtin_amdgcn_wmma_f32_16x16x4_f32(
                    false, a, false, b, (short)0, acc[tc], false, false);
            }
        }

        // bias + store + per-column stats
        #pragma unroll
        for (int tc = 0; tc < 4; ++tc) {
            int   col = tc * 16 + mrow;
            float bb  = bias[col];
            float s = 0.0f, q = 0.0f;
            #pragma unroll
            for (int r = 0; r < 8; ++r) {   // C layout: reg r -> row r + 8*half
                float v = acc[tc][r] + bb;
                Y[(wave * 16 + half * 8 + r) * DIM + col] = v;
                s += v;
                q += v * v;
            }
            s += __shfl_xor(s, 16, 32);     // combine the two wave halves
            q += __shfl_xor(q, 16, 32);
            if (half == 0) {
                atomicAdd(&ls[col], s);
                atomicAdd(&ls[64 + col], q);
            }
        }
    }
    __syncthreads();
    if (t < 64) {
        atomicAdd(&st_sum[t], ls[t]);
        atomicAdd(&st_sq[t],  ls[64 + t]);
    }
}

// ---------------------------------------------------------------------------
// Fold BN stats into per-column scale/shift:  y_norm = y*scale + shift
// ---------------------------------------------------------------------------
__global__ void k_finalize(const float* __restrict__ st_sum,
                           const float* __restrict__ st_sq,
                           const float* __restrict__ g,
                           const float* __restrict__ b,
                           float* __restrict__ scale,
                           float* __restrict__ shift) {
    int c = threadIdx.x;
    if (c < DIM) {
        float m = st_sum[c] * (1.0f / NODES);
        float v = st_sq[c] * (1.0f / NODES) - m * m;   // biased variance
        v = fmaxf(v, 0.0f);
        float a = g[c] * rsqrtf(v + BN_EPS);
        scale[c] = a;
        shift[c] = b[c] - m * a;
    }
}

// ---------------------------------------------------------------------------
// out = relu(y*scale + shift)   (final BN+ReLU of a layer)
// ---------------------------------------------------------------------------
__global__ __launch_bounds__(256) void k_apply(const float* __restrict__ y,
                                               const float* __restrict__ scale,
                                               const float* __restrict__ shift,
                                               float* __restrict__ out, int n) {
    int i = blockIdx.x * 256 + threadIdx.x;
    if (i >= n) return;
    int c = i & (DIM - 1);
    out[i] = fmaxf(y[i] * scale[c] + shift[c], 0.0f);
}

// ---------------------------------------------------------------------------
extern "C" void kernel_launch(void* const* d_in, const int* in_sizes, int n_in,
                              void* d_out, int out_size, void* d_ws, size_t ws_size,
                              hipStream_t stream) {
    const float* feat   = (const float*)d_in[0];
    const float* edge_w = (const float*)d_in[1];
    const float* W1     = (const float*)d_in[2];
    const float* b1     = (const float*)d_in[3];
    const float* bn1_g  = (const float*)d_in[4];
    const float* bn1_b  = (const float*)d_in[5];
    const float* W2     = (const float*)d_in[6];
    const float* b2     = (const float*)d_in[7];
    const float* bn_g   = (const float*)d_in[8];
    const float* bn_b   = (const float*)d_in[9];
    const float* eps    = (const float*)d_in[10];
    const int*   esrc   = (const int*)d_in[11];
    const int*   edst   = (const int*)d_in[12];
    const int E = in_sizes[1];

    const size_t ND = (size_t)NODES * DIM;
    float* ws  = (float*)d_ws;
    float* agg = ws;            // [N,64] aggregation accumulator / GEMM1 input
    float* y1  = ws + ND;       // [N,64] pre-BN1 activations
    float* y2  = ws + 2 * ND;   // [N,64] pre-BN2 activations
    float* hbf = ws + 3 * ND;   // [N,64] layer output
    float* st  = ws + 4 * ND;   // 256: sum1, sq1, sum2, sq2
    float* ab  = st + 256;      // 256: scale1, shift1, scale2, shift2

    const int n4          = (int)(ND / 4);
    const int row_tiles   = NODES / 16;                      // 6250
    const int gemm_blocks = (row_tiles + 7) / 8;
    const int scat_blocks = (int)(((size_t)E * 32 + 255) / 256);

    const float* hcur = feat;
    for (int l = 0; l < NLAYERS; ++l) {
        k_combine<<<(n4 + 255) / 256, 256, 0, stream>>>(hcur, eps, l, agg, n4);
        k_scatter<<<scat_blocks, 256, 0, stream>>>(hcur, edge_w, esrc, edst, agg, E);
        k_zero<<<1, 256, 0, stream>>>(st, 256);
        k_gemm_wmma<false><<<gemm_blocks, 256, 0, stream>>>(
            agg, W1 + l * 4096, b1 + l * 64, nullptr, nullptr,
            y1, st, st + 64, row_tiles);
        k_finalize<<<1, 64, 0, stream>>>(st, st + 64, bn1_g + l * 64, bn1_b + l * 64,
                                         ab, ab + 64);
        k_gemm_wmma<true><<<gemm_blocks, 256, 0, stream>>>(
            y1, W2 + l * 4096, b2 + l * 64, ab, ab + 64,
            y2, st + 128, st + 192, row_tiles);
        k_finalize<<<1, 64, 0, stream>>>(st + 128, st + 192, bn_g + l * 64, bn_b + l * 64,
                                         ab + 128, ab + 192);
        float* hnext = (l == NLAYERS - 1) ? (float*)d_out : hbf;
        k_apply<<<((int)ND + 255) / 256, 256, 0, stream>>>(y2, ab + 128, ab + 192,
                                                           hnext, (int)ND);
        hcur = hnext;
    }
}